// Transformer_34522947125817
// MI455X (gfx1250) — compile-verified
//
#include <hip/hip_runtime.h>
#include <math.h>

#define T_FRAMES 16384
#define FDIM     2048
#define COUT     7
#define LEN_Q    30
#define DMODEL   256   // N_HEADS * D_K
#define NHEADS   4
#define DK       64
#define DFF      64

#define KSPLIT   4
#define KCHUNK   (FDIM / KSPLIT)     // 512

typedef __attribute__((ext_vector_type(2))) float v2f;
typedef __attribute__((ext_vector_type(8))) float v8f;

// ---------------------------------------------------------------------------
// Kernel A0: zero-pad W_fc (7 x 2048) to Wpad (16 x 2048) so the WMMA hot
// loop needs no per-iteration mask multiply.
// ---------------------------------------------------------------------------
__global__ __launch_bounds__(256) void pad_w(
    const float* __restrict__ W, float* __restrict__ Wpad)
{
    const int i = blockIdx.x * 256 + threadIdx.x;    // 16*2048 elements
    const int row = i / FDIM;
    Wpad[i] = (row < COUT) ? W[i] : 0.0f;
}

// ---------------------------------------------------------------------------
// Kernel A1: split-K partial GEMM via V_WMMA_F32_16X16X4_F32.
// 4096 waves (1024 tiles x 4 K-chunks), each wave: 128 WMMA over 512 K.
// Unroll 16 -> 32 b64 loads in flight per wave for HBM latency hiding.
// A-frag (16x4 f32): lane L -> row = L%16, VGPR0/1 = K + (L/16)*2 + {0,1}
// B-frag (4x16 f32): lane L -> col = L%16, same K striping (padded rows = 0).
// D (16x16 f32):     lane L, vgpr r -> M = r + (L/16)*8, N = L%16
// ---------------------------------------------------------------------------
__global__ __launch_bounds__(256) void fc_partial_wmma(
    const float* __restrict__ A,      // (T, 2048)
    const float* __restrict__ Wpad,   // (16, 2048) zero-padded
    float* __restrict__ part)         // (KSPLIT, T, 8)
{
    const int wave  = threadIdx.x >> 5;
    const int lane  = threadIdx.x & 31;
    const int gw    = blockIdx.x * 8 + wave;         // 512 blocks * 8 = 4096
    const int tile  = gw >> 2;                       // 1024 tiles of 16 rows
    const int chunk = gw & 3;                        // K chunk
    const int row0  = tile * 16;
    const int m     = lane & 15;
    const int n     = lane & 15;
    const int kofs  = (lane >> 4) * 2;               // 0 or 2

    const float* arow = A    + (size_t)(row0 + m) * FDIM + chunk * KCHUNK + kofs;
    const float* brow = Wpad + (size_t)n          * FDIM + chunk * KCHUNK + kofs;

    v8f acc = {0.f, 0.f, 0.f, 0.f, 0.f, 0.f, 0.f, 0.f};

#pragma unroll 16
    for (int k = 0; k < KCHUNK; k += 4) {
        v2f a = *(const v2f*)(arow + k);
        v2f b = *(const v2f*)(brow + k);
        acc = __builtin_amdgcn_wmma_f32_16x16x4_f32(
            /*neg_a=*/false, a, /*neg_b=*/false, b,
            /*c_mod=*/(short)0, acc, /*reuse_a=*/false, /*reuse_b=*/false);
    }

    if (n < COUT) {
        const int mbase = row0 + (lane >> 4) * 8;
        float* pbase = part + (size_t)chunk * T_FRAMES * 8;
#pragma unroll
        for (int r = 0; r < 8; ++r)
            pbase[(size_t)(mbase + r) * 8 + n] = acc[r];
    }
}

// ---------------------------------------------------------------------------
// Kernel A2: reduce split-K partials + tanh -> feas (T x 8, col 7 zeroed).
// ---------------------------------------------------------------------------
__global__ __launch_bounds__(256) void reduce_tanh(
    const float* __restrict__ part,   // (KSPLIT, T, 8)
    float* __restrict__ feas)         // (T, 8)
{
    const int i = blockIdx.x * 256 + threadIdx.x;    // T*8 elements
    const int c = i & 7;
    float s = 0.f;
    if (c < COUT) {
#pragma unroll
        for (int ch = 0; ch < KSPLIT; ++ch)
            s += part[(size_t)ch * T_FRAMES * 8 + i];
        s = tanhf(s);
    }
    feas[i] = s;
}

// ---------------------------------------------------------------------------
// Kernel B: per-frame K/V projections (hoisted out of the sliding windows).
// ---------------------------------------------------------------------------
__global__ __launch_bounds__(256) void kv_proj(
    const float* __restrict__ x,      // (7, T)
    const float* __restrict__ Wk,     // (256, 7)
    const float* __restrict__ Wv,     // (256, 7)
    float* __restrict__ kp,           // (T, 256)
    float* __restrict__ vp)           // (T, 256)
{
    const int t = blockIdx.x;
    const int m = threadIdx.x;
    float f[COUT];
#pragma unroll
    for (int c = 0; c < COUT; ++c) f[c] = x[(size_t)c * T_FRAMES + t];
    float sk = 0.f, sv = 0.f;
#pragma unroll
    for (int c = 0; c < COUT; ++c) {
        sk += f[c] * Wk[m * COUT + c];
        sv += f[c] * Wv[m * COUT + c];
    }
    kp[(size_t)t * DMODEL + m] = sk;
    vp[(size_t)t * DMODEL + m] = sv;
}

// ---------------------------------------------------------------------------
// Kernel C: per-frame attention + FFN. 8 frames per 256-thread block (one
// wave32 per frame). The 37 K-rows covering the block's 8 overlapping
// windows are staged once into LDS (rows padded to 257 floats -> the
// stride-257 window reads in phase 2 are bank-conflict free).
// ---------------------------------------------------------------------------
#define KROWS 37          // [t0-29 .. t0+7]
#define KPAD  257

__global__ __launch_bounds__(256) void attn_ffn(
    const float* __restrict__ feas,   // (T, 8)
    const float* __restrict__ kp,     // (T, 256)
    const float* __restrict__ vp,     // (T, 256)
    const float* __restrict__ Wq,     // (256, 7)
    const float* __restrict__ Wo,     // (7, 256)
    const float* __restrict__ g1, const float* __restrict__ b1,
    const float* __restrict__ Wf1,    // (64, 7)
    const float* __restrict__ Wf2,    // (7, 64)
    const float* __restrict__ g2, const float* __restrict__ b2,
    float* __restrict__ out)          // (T, 7)
{
    __shared__ float s_k[KROWS * KPAD];       // staged K rows      (~38 KB)
    __shared__ float s_q[8][DMODEL];          // q, then ctx        (8 KB)
    __shared__ float s_attn[8][NHEADS * 32];  // attn, then hidden  (4 KB)
    __shared__ float s_misc[8][32];           // [0..6]=o  [8..14]=onorm [16..22]=y

    const int w    = threadIdx.x >> 5;
    const int lane = threadIdx.x & 31;
    const int t0   = blockIdx.x * 8;
    const int t    = t0 + w;

    // stage K rows for the whole block
    for (int i = threadIdx.x; i < KROWS * DMODEL; i += 256) {
        const int r = i >> 8, col = i & 255;
        const int g = t0 - (LEN_Q - 1) + r;
        if (g >= 0) s_k[r * KPAD + col] = kp[(size_t)g * DMODEL + col];
    }

    float fv[COUT];
#pragma unroll
    for (int c = 0; c < COUT; ++c) fv[c] = feas[(size_t)t * 8 + c];

    // phase 1: q = feas @ Wq.T  (256 values per frame, 8 per lane)
#pragma unroll
    for (int i = 0; i < 8; ++i) {
        const int mm = lane + 32 * i;
        float s = 0.f;
#pragma unroll
        for (int c = 0; c < COUT; ++c) s += fv[c] * Wq[mm * COUT + c];
        s_q[w][mm] = s;
    }
    __syncthreads();

    // phase 2: scores + softmax (lane j = window position)
    const int  j   = lane;
    const int  idx = t - (LEN_Q - 1) + j;
    const bool act = (j < LEN_Q);
    const bool inr = act && (idx >= 0);
    float sc[NHEADS];
#pragma unroll
    for (int h = 0; h < NHEADS; ++h) {
        float s = 0.f;
        if (inr) {
            const float* kr = &s_k[(w + j) * KPAD + h * DK];
#pragma unroll 8
            for (int d = 0; d < DK; ++d) s += s_q[w][h * DK + d] * kr[d];
            s *= 0.125f;                          // 1/sqrt(64)
        }
        sc[h] = act ? s : -3.402823466e38f;
    }
#pragma unroll
    for (int h = 0; h < NHEADS; ++h) {
        float mx = sc[h];
        for (int off = 16; off > 0; off >>= 1)
            mx = fmaxf(mx, __shfl_xor(mx, off, 32));
        float e = act ? __expf(sc[h] - mx) : 0.f;
        float ss = e;
        for (int off = 16; off > 0; off >>= 1)
            ss += __shfl_xor(ss, off, 32);
        s_attn[w][h * 32 + j] = e / ss;
    }
    __syncthreads();

    // phase 3: ctx = attn @ V (lane = dim; coalesced V reads, L2-resident)
    const int jstart = (t >= LEN_Q - 1) ? 0 : (LEN_Q - 1 - t);
#pragma unroll
    for (int h = 0; h < NHEADS; ++h) {
        float a0 = 0.f, a1 = 0.f;
        for (int jj = jstart; jj < LEN_Q; ++jj) {
            const float aw = s_attn[w][h * 32 + jj];
            const float* vr = vp + (size_t)(t - (LEN_Q - 1) + jj) * DMODEL + h * DK;
            a0 += aw * vr[lane];
            a1 += aw * vr[lane + 32];
        }
        s_q[w][h * DK + lane]      = a0;   // reuse s_q as ctx
        s_q[w][h * DK + 32 + lane] = a1;
    }
    __syncthreads();

    // phase 4: o = ctx @ Wo.T + feas; LayerNorm over C=7
    float oc = 0.f;
    if (lane < COUT) {
        for (int mm = 0; mm < DMODEL; ++mm) oc += s_q[w][mm] * Wo[lane * DMODEL + mm];
        oc += fv[lane];
        s_misc[w][lane] = oc;
    }
    __syncthreads();
    {
        float mu = 0.f;
#pragma unroll
        for (int c = 0; c < COUT; ++c) mu += s_misc[w][c];
        mu *= (1.0f / COUT);
        float var = 0.f;
#pragma unroll
        for (int c = 0; c < COUT; ++c) { float d = s_misc[w][c] - mu; var += d * d; }
        var *= (1.0f / COUT);
        const float rstd = rsqrtf(var + 1e-5f);
        if (lane < COUT)
            s_misc[w][8 + lane] = (oc - mu) * rstd * g1[lane] + b1[lane];
    }
    __syncthreads();

    // phase 5: hidden = relu(onorm @ Wff1.T), 64 values (reuse s_attn)
    {
        float on[COUT];
#pragma unroll
        for (int c = 0; c < COUT; ++c) on[c] = s_misc[w][8 + c];
#pragma unroll
        for (int p = 0; p < 2; ++p) {
            const int f = lane + 32 * p;
            float hs = 0.f;
#pragma unroll
            for (int c = 0; c < COUT; ++c) hs += on[c] * Wf1[f * COUT + c];
            s_attn[w][f] = fmaxf(hs, 0.f);
        }
    }
    __syncthreads();

    // phase 6: y = hidden @ Wff2.T + onorm; LayerNorm; store
    float yc = 0.f;
    if (lane < COUT) {
        for (int f = 0; f < DFF; ++f) yc += s_attn[w][f] * Wf2[lane * DFF + f];
        yc += s_misc[w][8 + lane];
        s_misc[w][16 + lane] = yc;
    }
    __syncthreads();
    {
        float mu = 0.f;
#pragma unroll
        for (int c = 0; c < COUT; ++c) mu += s_misc[w][16 + c];
        mu *= (1.0f / COUT);
        float var = 0.f;
#pragma unroll
        for (int c = 0; c < COUT; ++c) { float d = s_misc[w][16 + c] - mu; var += d * d; }
        var *= (1.0f / COUT);
        const float rstd = rsqrtf(var + 1e-5f);
        if (lane < COUT)
            out[(size_t)t * COUT + lane] = (yc - mu) * rstd * g2[lane] + b2[lane];
    }
}

// ---------------------------------------------------------------------------
extern "C" void kernel_launch(void* const* d_in, const int* in_sizes, int n_in,
                              void* d_out, int out_size, void* d_ws, size_t ws_size,
                              hipStream_t stream) {
    const float* x   = (const float*)d_in[0];   // (1,7,T)
    const float* lf  = (const float*)d_in[1];   // (1,T,2048)
    const float* Wfc = (const float*)d_in[2];   // (7,2048)
    const float* Wq  = (const float*)d_in[3];   // (256,7)
    const float* Wk  = (const float*)d_in[4];   // (256,7)
    const float* Wv  = (const float*)d_in[5];   // (256,7)
    const float* Wo  = (const float*)d_in[6];   // (7,256)
    const float* g1  = (const float*)d_in[7];
    const float* b1  = (const float*)d_in[8];
    const float* Wf1 = (const float*)d_in[9];   // (64,7)
    const float* Wf2 = (const float*)d_in[10];  // (7,64)
    const float* g2  = (const float*)d_in[11];
    const float* b2  = (const float*)d_in[12];
    float* out = (float*)d_out;

    float* feas = (float*)d_ws;                            // T x 8            (512 KB)
    float* kp   = feas + (size_t)T_FRAMES * 8;             // T x 256          (16 MB)
    float* vp   = kp   + (size_t)T_FRAMES * DMODEL;        // T x 256          (16 MB)
    float* part = vp   + (size_t)T_FRAMES * DMODEL;        // KSPLIT x T x 8   (2 MB)
    float* Wpad = part + (size_t)KSPLIT * T_FRAMES * 8;    // 16 x 2048        (128 KB)

    pad_w<<<(16 * FDIM) / 256, 256, 0, stream>>>(Wfc, Wpad);
    fc_partial_wmma<<<(T_FRAMES / 16) * KSPLIT / 8, 256, 0, stream>>>(lf, Wpad, part);
    reduce_tanh<<<(T_FRAMES * 8) / 256, 256, 0, stream>>>(part, feas);
    kv_proj<<<T_FRAMES, 256, 0, stream>>>(x, Wk, Wv, kp, vp);
    attn_ffn<<<T_FRAMES / 8, 256, 0, stream>>>(feas, kp, vp, Wq, Wo,
                                               g1, b1, Wf1, Wf2, g2, b2, out);
}